// CrossEntropyBoundSmoothLoss_60052232732847
// MI455X (gfx1250) — compile-verified
//
#include <hip/hip_runtime.h>

// CrossEntropyBoundSmoothLoss for MI455X (gfx1250).
// Memory-bound: 105 MB logits read exactly once (~4.5us @ 23.3 TB/s).
//  - Coalesced async global->LDS staging (GLOBAL_LOAD_ASYNC_TO_LDS_B128,
//    ASYNCcnt) of a 64-row tile per block; padded LDS stride 204 floats
//    makes the WMMA-A-layout ds_load_b64 reads bank-conflict-free.
//  - v_wmma_f32_16x16x4_f32 with B = ones performs the per-row sum(exp).
//  - Sparse smoothed-target dot (<=6 gathers/row) from the LDS tile.

typedef float v2f __attribute__((ext_vector_type(2)));
typedef float v8f __attribute__((ext_vector_type(8)));

#define SS   2048      // seq_len
#define LL   200       // label count
#define DD   2         // window radius
#define EE   0.1f      // smoothing mass
#define RPB  64        // rows per block
#define LSTR 204       // padded LDS row stride (floats): 16B-aligned, 64-bank clean

__global__ void cebs_zero(float* out, int n) {
    int i = blockIdx.x * blockDim.x + threadIdx.x;
    if (i < n) out[i] = 0.0f;
}

__global__ __launch_bounds__(256)
void cebs_kernel(const float* __restrict__ logits,
                 const int*   __restrict__ lab,
                 float* __restrict__ out, int nrows)
{
    __shared__ float sTile[RPB * LSTR];   // 52224 B
    __shared__ float sMax[8 * 16];        // [tile*2+half][row]
    __shared__ float sSum[8 * 16];

    const int tid       = threadIdx.x;
    const int lane      = tid & 31;
    const int wid       = tid >> 5;
    const int blockRow0 = blockIdx.x * RPB;

    // ---- stage 64x200 f32 rows (contiguous in global) into LDS, async ------
    {
        int rowsHere = nrows - blockRow0;
        if (rowsHere > RPB) rowsHere = RPB;
        const int nvec = rowsHere * (LL / 4);             // float4 count (<=3200)
        const unsigned long long gbase =
            (unsigned long long)(const void*)(logits + (size_t)blockRow0 * LL);
        for (int i = tid; i < nvec; i += 256) {
            const int row = i / (LL / 4);
            const int c4  = i - row * (LL / 4);
            const unsigned lds =
                (unsigned)(unsigned long long)(const void*)(&sTile[row * LSTR + c4 * 4]);
            const unsigned long long ga = gbase + (unsigned long long)i * 16ull;
            // VDST = LDS byte address, VADDR = 64-bit global address (ASYNCcnt)
            asm volatile("global_load_async_to_lds_b128 %0, %1, off"
                         :: "v"(lds), "v"(ga) : "memory");
        }
        asm volatile("s_wait_asynccnt 0x0" ::: "memory");
        __syncthreads();
    }

    // ---- per-wave assignment: tile t (16 rows) x column half h (100 cols) --
    const int t    = wid & 3;
    const int h    = wid >> 2;
    const int m    = lane & 15;                 // owned matrix row in tile
    const int koff = (lane >> 4) << 1;          // A-layout: 0 / 2
    const float* __restrict__ rowp = &sTile[(t * 16 + m) * LSTR];
    const int cbase = h * 100 + koff;

    // pass 1: partial row max over this wave's half (conflict-free ds_load_b64)
    float mx = -3.0e38f;
    #pragma unroll 5
    for (int k = 0; k < 100; k += 4) {
        float2 v = *(const float2*)(rowp + cbase + k);
        mx = fmaxf(mx, fmaxf(v.x, v.y));
    }
    mx = fmaxf(mx, __shfl_xor(mx, 16));         // combine lane pair
    if (lane < 16) sMax[(t * 2 + h) * 16 + m] = mx;
    __syncthreads();
    mx = fmaxf(sMax[(t * 2 + 0) * 16 + m], sMax[(t * 2 + 1) * 16 + m]);

    // pass 2: sum(exp(x-mx)) via WMMA: D = A x ones + C accumulates row sums
    v8f c = {0.f, 0.f, 0.f, 0.f, 0.f, 0.f, 0.f, 0.f};
    v2f ones; ones[0] = 1.0f; ones[1] = 1.0f;
    #pragma unroll 5
    for (int k = 0; k < 100; k += 4) {
        float2 v = *(const float2*)(rowp + cbase + k);
        v2f a;
        a[0] = __expf(v.x - mx);
        a[1] = __expf(v.y - mx);
        c = __builtin_amdgcn_wmma_f32_16x16x4_f32(false, a, false, ones,
                                                  (short)0, c, false, false);
    }
    // C/D layout: vgpr j, lanes 0-15 -> M=j; lanes 16-31 -> M=j+8 (any N)
    if (lane == 0) {
        #pragma unroll
        for (int j = 0; j < 8; ++j) sSum[(t * 2 + h) * 16 + j] = c[j];
    } else if (lane == 16) {
        #pragma unroll
        for (int j = 0; j < 8; ++j) sSum[(t * 2 + h) * 16 + 8 + j] = c[j];
    }
    __syncthreads();

    // ---- sparse smoothed-target dot: threads 0..63 own one row each --------
    float loss = 0.0f;
    if (tid < RPB) {
        const int i = tid;
        const int r = blockRow0 + i;
        if (r < nrows) {
            const int tt = i >> 4, mm = i & 15;
            const float fm = fmaxf(sMax[(tt * 2 + 0) * 16 + mm],
                                   sMax[(tt * 2 + 1) * 16 + mm]);
            const float Z  = sSum[(tt * 2 + 0) * 16 + mm] +
                             sSum[(tt * 2 + 1) * 16 + mm];
            const float lse = fm + __logf(Z);
            const float* __restrict__ rp = &sTile[i * LSTR];

            const int s    = r & (SS - 1);
            const int bofs = r - s;                     // b * SS
            const int lo = (s - DD < 0) ? 0 : s - DD;
            const int hi = (s + DD > SS - 1) ? SS - 1 : s + DD;
            float tdot = 0.0f, tsum = 0.0f;
            int seen[2 * DD + 1];
            int ns = 0;
            // walk window top-down: first hit per boundary label is its last_t
            for (int tw = hi; tw >= lo; --tw) {
                const int l = lab[bofs + tw];
                if (l % 10 != 0) continue;              // boundary ids 0,10,...,190
                bool dup = false;
                for (int j = 0; j < ns; ++j) dup |= (seen[j] == l);
                if (dup) continue;
                seen[ns++] = l;
                const int wlo = (tw - DD < 0) ? 0 : tw - DD;
                const int whi = (tw + DD > SS - 1) ? SS - 1 : tw + DD;
                const int w   = (whi - wlo) < 1 ? 1 : (whi - wlo);
                const float wt = (s == tw) ? (1.0f - EE) : (EE / (float)w);
                tdot += wt * rp[l];
                tsum += wt;
            }
            const int l0 = lab[r];
            if (l0 % 10 != 0) { tdot += rp[l0]; tsum += 1.0f; }
            // -sum(target * logprob) = tsum*(max+logZ) - sum(target*logit)
            loss = tsum * lse - tdot;
        }
    }

    // ---- reduce: waves 0-1 hold all contributions; 1 atomic per wave -------
    if (wid < 2) {
        #pragma unroll
        for (int o = 16; o > 0; o >>= 1) loss += __shfl_xor(loss, o);
        if (lane == 0) atomicAdd(out, loss * (1.0f / (float)nrows));
    }
}

extern "C" void kernel_launch(void* const* d_in, const int* in_sizes, int n_in,
                              void* d_out, int out_size, void* d_ws, size_t ws_size,
                              hipStream_t stream) {
    const float* logits = (const float*)d_in[0];
    const int*   lab    = (const int*)d_in[1];
    float*       out    = (float*)d_out;
    const int nrows = in_sizes[1];                  // B*S = 131072

    cebs_zero<<<1, 64, 0, stream>>>(out, out_size);

    const int blocks = (nrows + RPB - 1) / RPB;     // 2048 for B=64,S=2048
    cebs_kernel<<<blocks, 256, 0, stream>>>(logits, lab, out, nrows);
}